// MultiHeadAttention_13950053778299
// MI455X (gfx1250) — compile-verified
//
#include <hip/hip_runtime.h>
#include <hip/hip_bf16.h>

// MI455X / gfx1250, wave32. Memory-bound fused decode-MHA:
//  - Q/K/V projections algebraically folded around the length-1 query so that
//    key (256MB) and value (256MB) are each streamed exactly once from HBM
//    (~550MB total @ 23.3TB/s -> ~23us floor; compute collapses to ~6 GFLOP).
//  - All GEMM-like work uses V_WMMA_F32_16X16X4_F32 (fp32 A/B, fp32 acc):
//    bandwidth-bound, so fp32 matrix rate is free and numerics match fp32 ref.
//  - The two streaming kernels use double-buffered GLOBAL_LOAD_ASYNC_TO_LDS_B128
//    (ASYNCcnt-tracked, s_wait_asynccnt) so HBM streaming overlaps WMMA.

typedef __attribute__((ext_vector_type(2))) float v2f;
typedef __attribute__((ext_vector_type(4))) float v4f;
typedef __attribute__((ext_vector_type(8))) float v8f;

#define NBATCH 256
#define SEQL   256
#define HDIM   1024
#define NHEAD  16
#define DHEAD  64
#define SCALE_F 0.5f      // (DHEAD // NHEAD) ** -0.5 = 4^-0.5
#define LN_EPS_F 1e-5f

__device__ __forceinline__ v8f wmma4(v2f a, v2f b, v8f c) {
  // D(16x16,f32) = A(16x4,f32) x B(4x16,f32) + C
  return __builtin_amdgcn_wmma_f32_16x16x4_f32(false, a, false, b, (short)0, c,
                                               false, false);
}

// Async 16B copy global -> LDS. Generic pointers to __shared__ carry the LDS
// byte offset in their low 32 bits (LDS aperture lives entirely in addr[63:32]).
__device__ __forceinline__ void async_b128(const float* gsrc, const float* ldst) {
  unsigned l = (unsigned)(unsigned long long)ldst;
  asm volatile("global_load_async_to_lds_b128 %0, %1, off"
               :: "v"(l), "v"(gsrc) : "memory");
}

// ---------------------------------------------------------------------------
// C[m][n] = sum_k A[m*K+k] * W[n*K+k] + bias[n] (+ resid[m*N+n])
// One 16x16 tile per wave; lane layout per CDNA5 ISA 7.12.2:
//   A: row = lane&15, k-pair = 2*(lane>>4);  B/D: col = lane&15.
__global__ void k_gemm_awt(const float* __restrict__ A, const float* __restrict__ W,
                           const float* __restrict__ bias, const float* __restrict__ resid,
                           float* __restrict__ out, int M, int N, int K) {
  const int lane = threadIdx.x & 31, wave = threadIdx.x >> 5;
  const int ntiles = N >> 4;
  const int t = blockIdx.x * (blockDim.x >> 5) + wave;
  if (t >= (M >> 4) * ntiles) return;
  const int m0 = (t / ntiles) << 4, n0 = (t % ntiles) << 4;
  const int r = lane & 15, half = lane >> 4;
  const float* ap = A + (size_t)(m0 + r) * K + 2 * half;
  const float* bp = W + (size_t)(n0 + r) * K + 2 * half;   // B[k][n] = W[n][k]
  v8f acc = {};
  for (int k0 = 0; k0 < K; k0 += 64) {
    __builtin_prefetch(bp + k0 + 512, 0, 1);               // global_prefetch_b8
    #pragma unroll
    for (int k = k0; k < k0 + 64; k += 4) {
      v2f a = *(const v2f*)(ap + k);
      v2f b = *(const v2f*)(bp + k);
      acc = wmma4(a, b, acc);
    }
  }
  const int n = n0 + r;
  const float bn = bias[n];
  #pragma unroll
  for (int i = 0; i < 8; ++i) {
    const int m = m0 + i + 8 * half;
    float v = acc[i] + bn;
    if (resid) v += resid[(size_t)m * N + n];
    out[(size_t)m * N + n] = v;
  }
}

// wq_eff[b][h][j] = sum_d q[b][h*64+d] * Wk[h*64+d][j]   (per-head q^T Wk fold)
__global__ void k_wqeff(const float* __restrict__ q, const float* __restrict__ Wk,
                        float* __restrict__ wq) {
  const int lane = threadIdx.x & 31, wave = threadIdx.x >> 5;
  const int gt = blockIdx.x * (blockDim.x >> 5) + wave;   // 16 heads * 16 mt * 64 nt
  const int h = gt >> 10;
  const int rem = gt & 1023;
  const int m0 = (rem >> 6) << 4;
  const int n0 = (rem & 63) << 4;
  const int r = lane & 15, half = lane >> 4;
  const float* ap = q + (size_t)(m0 + r) * HDIM + h * DHEAD + 2 * half;
  const float* bp = Wk + (size_t)(h * DHEAD + 2 * half) * HDIM + n0 + r;
  v8f acc = {};
  for (int k = 0; k < DHEAD; k += 4) {
    v2f a = *(const v2f*)(ap + k);
    v2f b;
    b.x = bp[(size_t)k * HDIM];
    b.y = bp[(size_t)(k + 1) * HDIM];
    acc = wmma4(a, b, acc);
  }
  #pragma unroll
  for (int i = 0; i < 8; ++i) {
    const int m = m0 + i + 8 * half;
    wq[((size_t)m * NHEAD + h) * HDIM + n0 + r] = acc[i];
  }
}

// qbk[b][h] = sum_d q[b][h*64+d] * bk[h*64+d]   (bias fold for scores)
__global__ void k_qbk(const float* __restrict__ q, const float* __restrict__ bk,
                      float* __restrict__ qbk) {
  const int t = blockIdx.x * blockDim.x + threadIdx.x;
  if (t >= NBATCH * NHEAD) return;
  const int b = t >> 4, h = t & 15;
  float s = 0.f;
  for (int d = 0; d < DHEAD; ++d)
    s += q[(size_t)b * HDIM + h * DHEAD + d] * bk[h * DHEAD + d];
  qbk[t] = s;
}

// ---------------------------------------------------------------------------
// Per batch b: scores[h][l] = SCALE*(key[b,l,:].wq_eff[b,h,:] + qbk[b,h]),
// mask, softmax over l, write attention. Streams key once through async-LDS
// double buffer; WMMA A operand from LDS, B operand (wq_eff) from LDS.
#define LWQ_STRIDE 1028   // conflict-free column reads (4r banks)
#define SC_STRIDE  17     // conflict-free strided column reads (17*lane banks)
#define KC         32     // k-chunk columns
#define KB_PITCH   34     // kbuf row pitch (34r mod 64 distinct, v2f-aligned)
__global__ void k_attn(const float* __restrict__ key, const float* __restrict__ wq,
                       const float* __restrict__ qbk, const unsigned char* __restrict__ mask,
                       float* __restrict__ attn /* [B*NH][L] */) {
  __shared__ float lwq[NHEAD * LWQ_STRIDE];       // 65792 B
  __shared__ float sc[SEQL * SC_STRIDE];          // 17408 B
  __shared__ float kbuf[2][SEQL * KB_PITCH];      // 2 x 34816 B
  const int b = blockIdx.x;
  const int tid = threadIdx.x;                    // 512 threads = 16 waves
  const float* keyb = key + (size_t)b * SEQL * HDIM;

  // Kick off chunk 0 immediately (4 x b128 per thread = 8KB/chunk-row-group).
  {
    #pragma unroll
    for (int j = 0; j < 4; ++j) {
      const int e = tid + 512 * j;                // 2048 16B segments / chunk
      const int row = e >> 3, c16 = e & 7;
      async_b128(keyb + (size_t)row * HDIM + c16 * 4,
                 &kbuf[0][row * KB_PITCH + c16 * 4]);
    }
  }
  // Stage wq_eff[b] (B operand) while chunk 0 is in flight.
  for (int idx = tid; idx < NHEAD * (HDIM / 4); idx += blockDim.x) {
    const int h = idx >> 8, k4 = idx & 255;
    v4f v = *(const v4f*)(wq + ((size_t)b * NHEAD + h) * HDIM + k4 * 4);
    *(v4f*)(&lwq[h * LWQ_STRIDE + k4 * 4]) = v;
  }

  const int lane = tid & 31, wave = tid >> 5;
  const int r = lane & 15, half = lane >> 4;
  const int m0 = wave << 4;                       // 16 key rows per wave
  v8f acc = {};
  const int NCHUNK = HDIM / KC;                   // 32
  for (int c = 0; c < NCHUNK; ++c) {
    if (c + 1 < NCHUNK) {                         // issue chunk c+1
      const int k0 = (c + 1) * KC;
      const int buf = (c + 1) & 1;
      #pragma unroll
      for (int j = 0; j < 4; ++j) {
        const int e = tid + 512 * j;
        const int row = e >> 3, c16 = e & 7;
        async_b128(keyb + (size_t)row * HDIM + k0 + c16 * 4,
                   &kbuf[buf][row * KB_PITCH + c16 * 4]);
      }
      asm volatile("s_wait_asynccnt 4" ::: "memory");  // chunk c landed
    } else {
      asm volatile("s_wait_asynccnt 0" ::: "memory");
    }
    __syncthreads();                              // chunk c visible block-wide
    const float* ap = &kbuf[c & 1][(m0 + r) * KB_PITCH + 2 * half];
    const float* bp = &lwq[r * LWQ_STRIDE + c * KC + 2 * half];
    #pragma unroll
    for (int k = 0; k < KC; k += 4)
      acc = wmma4(*(const v2f*)(ap + k), *(const v2f*)(bp + k), acc);
    __syncthreads();                              // done reading buf before reuse
  }

  const float qb = SCALE_F * qbk[b * NHEAD + r];  // n-dim == head == r
  #pragma unroll
  for (int i = 0; i < 8; ++i) {
    const int l = m0 + i + 8 * half;
    float v = acc[i] * SCALE_F + qb;
    if (mask[b * SEQL + l]) v = -__builtin_inff();  // mask==True -> -inf
    sc[l * SC_STRIDE + r] = v;
  }
  __syncthreads();

  // Softmax: wave w owns head h=w; wave32 shuffle reductions.
  const int h = wave;
  float mx = -__builtin_inff();
  #pragma unroll
  for (int j = 0; j < SEQL / 32; ++j)
    mx = fmaxf(mx, sc[(lane + 32 * j) * SC_STRIDE + h]);
  #pragma unroll
  for (int off = 16; off > 0; off >>= 1)
    mx = fmaxf(mx, __shfl_xor(mx, off, 32));
  float e[SEQL / 32], sum = 0.f;
  #pragma unroll
  for (int j = 0; j < SEQL / 32; ++j) {
    e[j] = __expf(sc[(lane + 32 * j) * SC_STRIDE + h] - mx);
    sum += e[j];
  }
  #pragma unroll
  for (int off = 16; off > 0; off >>= 1)
    sum += __shfl_xor(sum, off, 32);
  const float inv = 1.0f / sum;
  #pragma unroll
  for (int j = 0; j < SEQL / 32; ++j)
    attn[((size_t)b * NHEAD + h) * SEQL + lane + 32 * j] = e[j] * inv;
}

// ---------------------------------------------------------------------------
// vbar[b][h][j] = sum_l attn[b,h,l] * value[b,l,j].  Streams value once via
// async-LDS double buffer (16 l-rows x 1024 cols per chunk).
#define LA_STRIDE 260
#define KR        16      // l-rows per chunk
#define VB_PITCH  1040    // vbuf row pitch: half-wave row skew of 32 banks
__global__ void k_vbar(const float* __restrict__ value, const float* __restrict__ attn,
                       float* __restrict__ vbar) {
  __shared__ float la[NHEAD * LA_STRIDE];         // 16640 B
  __shared__ float vbuf[2][KR * VB_PITCH];        // 2 x 66560 B
  const int b = blockIdx.x;
  const int tid = threadIdx.x;                    // 512 threads = 16 waves
  const float* vb = value + (size_t)b * SEQL * HDIM;

  // Kick off chunk 0 (8 x b128 per thread; 4096 segments = 64KB).
  {
    #pragma unroll
    for (int j = 0; j < 8; ++j) {
      const int e = tid + 512 * j;
      const int row = e >> 8, c16 = e & 255;
      async_b128(vb + (size_t)row * HDIM + c16 * 4,
                 &vbuf[0][row * VB_PITCH + c16 * 4]);
    }
  }
  // Stage attn[b] (A operand, 16x256) while chunk 0 is in flight.
  for (int idx = tid; idx < NHEAD * (SEQL / 4); idx += blockDim.x) {
    const int h = idx >> 6, k4 = idx & 63;
    v4f v = *(const v4f*)(attn + ((size_t)b * NHEAD + h) * SEQL + k4 * 4);
    *(v4f*)(&la[h * LA_STRIDE + k4 * 4]) = v;
  }

  const int lane = tid & 31, wave = tid >> 5;
  const int r = lane & 15, half = lane >> 4;
  v8f acc[4] = {{}, {}, {}, {}};                  // 4 n-tiles per wave
  const int NCHUNK = SEQL / KR;                   // 16
  for (int c = 0; c < NCHUNK; ++c) {
    if (c + 1 < NCHUNK) {                         // issue chunk c+1
      const int k0 = (c + 1) * KR;
      const int buf = (c + 1) & 1;
      #pragma unroll
      for (int j = 0; j < 8; ++j) {
        const int e = tid + 512 * j;
        const int row = e >> 8, c16 = e & 255;
        async_b128(vb + (size_t)(k0 + row) * HDIM + c16 * 4,
                   &vbuf[buf][row * VB_PITCH + c16 * 4]);
      }
      asm volatile("s_wait_asynccnt 8" ::: "memory");  // chunk c landed
    } else {
      asm volatile("s_wait_asynccnt 0" ::: "memory");
    }
    __syncthreads();
    const float* bb = vbuf[c & 1];
    const float* ap = &la[r * LA_STRIDE + c * KR + 2 * half];  // A[h=r][k]
    #pragma unroll
    for (int j = 0; j < 4; ++j) {
      const int n0 = (wave * 4 + j) << 4;
      #pragma unroll
      for (int k = 0; k < KR; k += 4) {
        v2f a = *(const v2f*)(ap + k);
        v2f bv;
        bv.x = bb[(k + 2 * half) * VB_PITCH + n0 + r];       // B[k][n]=value
        bv.y = bb[(k + 2 * half + 1) * VB_PITCH + n0 + r];
        acc[j] = wmma4(a, bv, acc[j]);
      }
    }
    __syncthreads();
  }
  #pragma unroll
  for (int j = 0; j < 4; ++j) {
    const int n0 = (wave * 4 + j) << 4;
    #pragma unroll
    for (int i = 0; i < 8; ++i) {
      const int h = i + 8 * half;
      vbar[((size_t)b * NHEAD + h) * HDIM + n0 + r] = acc[j][i];
    }
  }
}

// context[b][h*64+n] = sum_j vbar[b][h][j] * Wv[h*64+n][j] + bv[h*64+n]
__global__ void k_ctx(const float* __restrict__ vbar, const float* __restrict__ Wv,
                      const float* __restrict__ bv, float* __restrict__ ctx) {
  const int lane = threadIdx.x & 31, wave = threadIdx.x >> 5;
  const int gt = blockIdx.x * (blockDim.x >> 5) + wave;  // 16 heads * 16 mt * 4 nt
  const int h = gt >> 6;
  const int rem = gt & 63;
  const int m0 = (rem >> 2) << 4;
  const int n0 = (rem & 3) << 4;
  const int r = lane & 15, half = lane >> 4;
  const float* ap = vbar + ((size_t)(m0 + r) * NHEAD + h) * HDIM + 2 * half;
  const float* bp = Wv + (size_t)(h * DHEAD + n0 + r) * HDIM + 2 * half;
  v8f acc = {};
  for (int k = 0; k < HDIM; k += 4) {
    v2f a = *(const v2f*)(ap + k);
    v2f bw = *(const v2f*)(bp + k);
    acc = wmma4(a, bw, acc);
  }
  const float bias = bv[h * DHEAD + n0 + r];
  #pragma unroll
  for (int i = 0; i < 8; ++i) {
    const int m = m0 + i + 8 * half;
    ctx[(size_t)m * HDIM + h * DHEAD + n0 + r] = acc[i] + bias;
  }
}

// LayerNorm over H per batch row.
__global__ void k_ln(const float* __restrict__ x, const float* __restrict__ gamma,
                     const float* __restrict__ beta, float* __restrict__ out) {
  __shared__ float red[256];
  const int b = blockIdx.x, tid = threadIdx.x;
  const float* xb = x + (size_t)b * HDIM;
  float v[4], s = 0.f;
  #pragma unroll
  for (int j = 0; j < 4; ++j) { v[j] = xb[tid + 256 * j]; s += v[j]; }
  red[tid] = s; __syncthreads();
  for (int off = 128; off > 0; off >>= 1) {
    if (tid < off) red[tid] += red[tid + off];
    __syncthreads();
  }
  const float mu = red[0] * (1.0f / HDIM);
  __syncthreads();
  float s2 = 0.f;
  #pragma unroll
  for (int j = 0; j < 4; ++j) { const float d = v[j] - mu; s2 += d * d; }
  red[tid] = s2; __syncthreads();
  for (int off = 128; off > 0; off >>= 1) {
    if (tid < off) red[tid] += red[tid + off];
    __syncthreads();
  }
  const float rinv = rsqrtf(red[0] * (1.0f / HDIM) + LN_EPS_F);
  #pragma unroll
  for (int j = 0; j < 4; ++j) {
    const int n = tid + 256 * j;
    out[(size_t)b * HDIM + n] = (v[j] - mu) * rinv * gamma[n] + beta[n];
  }
}

extern "C" void kernel_launch(void* const* d_in, const int* in_sizes, int n_in,
                              void* d_out, int out_size, void* d_ws, size_t ws_size,
                              hipStream_t stream) {
  const float* key   = (const float*)d_in[0];   // (B,L,H)
  const float* value = (const float*)d_in[1];   // (B,L,H)
  const float* query = (const float*)d_in[2];   // (B,1,H)
  const unsigned char* mask = (const unsigned char*)d_in[3]; // (B,1,L) bool (1B)
  const float* Wq = (const float*)d_in[4];
  const float* bq = (const float*)d_in[5];
  const float* Wk = (const float*)d_in[6];
  const float* bk = (const float*)d_in[7];
  const float* Wv = (const float*)d_in[8];
  const float* bv = (const float*)d_in[9];
  const float* Wf = (const float*)d_in[10];
  const float* bf = (const float*)d_in[11];
  const float* gamma = (const float*)d_in[12];
  const float* beta  = (const float*)d_in[13];

  float* out      = (float*)d_out;                  // (B,1,H)
  float* attn_out = out + NBATCH * HDIM;            // (B*NH,1,L)

  float* ws      = (float*)d_ws;                    // ~35 MB total
  float* ws_q    = ws;                              // B*H
  float* ws_wq   = ws_q   + NBATCH * HDIM;          // B*NH*H
  float* ws_qbk  = ws_wq  + NBATCH * NHEAD * HDIM;  // B*NH
  float* ws_vbar = ws_qbk + NBATCH * NHEAD;         // B*NH*H
  float* ws_ctx  = ws_vbar + NBATCH * NHEAD * HDIM; // B*H
  float* ws_x    = ws_ctx + NBATCH * HDIM;          // B*H

  // 1) q = query @ Wq^T + bq
  k_gemm_awt<<<128, 256, 0, stream>>>(query, Wq, bq, nullptr, ws_q, NBATCH, HDIM, HDIM);
  // 2) wq_eff = per-head q^T Wk fold
  k_wqeff<<<2048, 256, 0, stream>>>(ws_q, Wk, ws_wq);
  // 3) qbk fold
  k_qbk<<<16, 256, 0, stream>>>(ws_q, bk, ws_qbk);
  // 4) scores + mask + softmax (streams key async->LDS, 1 block per batch)
  k_attn<<<NBATCH, 512, 0, stream>>>(key, ws_wq, ws_qbk, mask, attn_out);
  // 5) vbar = attn @ value (streams value async->LDS, 1 block per batch)
  k_vbar<<<NBATCH, 512, 0, stream>>>(value, attn_out, ws_vbar);
  // 6) context = per-head vbar @ Wv_head^T + bv
  k_ctx<<<128, 256, 0, stream>>>(ws_vbar, Wv, bv, ws_ctx);
  // 7) x = context @ Wf^T + bf + query
  k_gemm_awt<<<128, 256, 0, stream>>>(ws_ctx, Wf, bf, query, ws_x, NBATCH, HDIM, HDIM);
  // 8) layernorm -> out
  k_ln<<<NBATCH, 256, 0, stream>>>(ws_x, gamma, beta, out);
}